// RecurrentNetTimeFixed_25503515803768
// MI455X (gfx1250) — compile-verified
//
#include <hip/hip_runtime.h>

typedef __attribute__((ext_vector_type(2))) float v2f;
typedef __attribute__((ext_vector_type(8))) float v8f;
typedef __attribute__((ext_vector_type(4))) unsigned int u32x4;
typedef __attribute__((ext_vector_type(8))) int i32x8;
typedef __attribute__((ext_vector_type(4))) int i32x4;

#define N_IN   128
#define N_OUT  64
#define N_HID  500
#define HPAD   512
#define BATCH  256
#define LENGTH 500
#define MROWS  (BATCH * LENGTH)   // 128000

#define WMMA_F32(a, b, c) \
  __builtin_amdgcn_wmma_f32_16x16x4_f32(false, (a), false, (b), (short)0, (c), false, false)

// ---------------------------------------------------------------------------
// Branchless wave-level 16x16 f32 tile of D = A @ B^T.
//   A: [M, K] row-major (lda) — caller guarantees rows m0..m0+15 valid
//   B: [N, K] row-major (ldb) — caller guarantees cols n0..n0+15 valid (pad!)
// ---------------------------------------------------------------------------
__device__ __forceinline__ v8f wmma_tile_nc(const float* __restrict__ A, long lda,
                                            const float* __restrict__ B, long ldb,
                                            int m0, int n0, int K) {
  const int lane = threadIdx.x & 31;
  const int half = lane >> 4;
  const int l16  = lane & 15;
  const float* Ap = A + (long)(m0 + l16) * lda + 2 * half;
  const float* Bp = B + (long)(n0 + l16) * ldb + 2 * half;
  v8f c = {};
#pragma unroll 5
  for (int k = 0; k < K; k += 4) {
    v2f a = *(const v2f*)(Ap + k);
    v2f b = *(const v2f*)(Bp + k);
    c = WMMA_F32(a, b, c);
  }
  return c;
}

// ---------------------------------------------------------------------------
// W_comb[n, k] = sum_j W_ih[n, j] * W_in[j, k]   (rows >= 500 come out zero)
// ---------------------------------------------------------------------------
__global__ void kCombW(const float* __restrict__ Wih, const float* __restrict__ Win,
                       float* __restrict__ Wc) {
  const int wave  = blockIdx.x * 4 + (threadIdx.x >> 5);
  const int mtile = wave >> 3;        // 32 mtiles (HPAD/16)
  const int ntile = wave & 7;         // 8  ntiles (N_IN/16)
  const int m0 = mtile * 16, n0 = ntile * 16;
  const int lane = threadIdx.x & 31;
  const int half = lane >> 4, l16 = lane & 15;
  const int row = m0 + l16;
  const float as = (row < N_HID) ? 1.f : 0.f;
  const float* Ap = Wih + (long)(row < N_HID ? row : 0) * N_HID + 2 * half;
  const float* Bt = Win + (long)(2 * half) * N_IN + (n0 + l16);
  v8f c = {};
#pragma unroll 5
  for (int k = 0; k < N_HID; k += 4) {
    v2f a = *(const v2f*)(Ap + k);
    a.x *= as; a.y *= as;
    v2f b;
    b.x = Bt[(long)k * N_IN];
    b.y = Bt[(long)(k + 1) * N_IN];
    c = WMMA_F32(a, b, c);
  }
#pragma unroll
  for (int j = 0; j < 8; ++j)
    Wc[(long)(m0 + j + 8 * half) * N_IN + (n0 + l16)] = c[j];
}

// bias_comb[n] = b_ih[n] + b_hh[n] + sum_j W_ih[n, j] * b_in[j]
__global__ void kBias(const float* __restrict__ Wih, const float* __restrict__ bin,
                      const float* __restrict__ bih, const float* __restrict__ bhh,
                      float* __restrict__ bc) {
  const int n = threadIdx.x;
  float s = 0.f;
  if (n < N_HID) {
    s = bih[n] + bhh[n];
    for (int j = 0; j < N_HID; ++j) s += Wih[(long)n * N_HID + j] * bin[j];
  }
  if (n < HPAD) bc[n] = s;
}

// Zero-pad W_hh [500,500] -> WhhP [512,500] so the step GEMM is branchless.
__global__ void kPadW(const float* __restrict__ Whh, float* __restrict__ WhhP) {
  long i = (long)blockIdx.x * blockDim.x + threadIdx.x;
  if (i < (long)HPAD * N_HID) {
    int r = (int)(i / N_HID);
    int k = (int)(i - (long)r * N_HID);
    WhhP[i] = (r < N_HID) ? Whh[(long)r * N_HID + k] : 0.f;
  }
}

// ---------------------------------------------------------------------------
// U[r, :] = X[r, :] @ W_comb^T + bias_comb     (W_comb zero-padded -> branchless)
// ---------------------------------------------------------------------------
__global__ void kU(const float* __restrict__ X, const float* __restrict__ Wc,
                   const float* __restrict__ bc, float* __restrict__ U) {
  const int wave  = blockIdx.x * 4 + (threadIdx.x >> 5);
  const int mtile = wave >> 5;        // 8000 mtiles
  const int ntile = wave & 31;        // 32 ntiles
  const int m0 = mtile * 16, n0 = ntile * 16;
  __builtin_prefetch(X + (long)m0 * N_IN, 0, 0);
  __builtin_prefetch(X + (long)(m0 + 8) * N_IN, 0, 0);
  v8f c = wmma_tile_nc(X, N_IN, Wc, N_IN, m0, n0, N_IN);
  const int lane = threadIdx.x & 31;
  const int half = lane >> 4, l16 = lane & 15;
#pragma unroll
  for (int j = 0; j < 8; ++j)
    U[(long)(m0 + j + 8 * half) * HPAD + (n0 + l16)] = c[j] + bc[n0 + l16];
}

// ---------------------------------------------------------------------------
// One recurrence step t. Block = 4 waves sharing one 16-col weight tile
// (64 batch rows x 16 hidden cols). The 16x500 f32 tile of padded W_hh is
// staged row-major in LDS by the Tensor Data Mover (wave 0 issues the D#),
// waited on before A-operand address setup; sync = s_wait_tensorcnt + barrier.
// ---------------------------------------------------------------------------
__global__ void kStep(const float* __restrict__ h0, float* __restrict__ hlist,
                      const float* __restrict__ U, const float* __restrict__ WhhP,
                      const float* __restrict__ alpha, int t) {
  __shared__ __align__(16) float ldsB[16 * N_HID];   // 16 rows x 500 f32 = 32000 B
  const int tid  = threadIdx.x;
  const int nblk = blockIdx.x & 31;       // 32 n tiles
  const int mblk = blockIdx.x >> 5;       // 4 m blocks of 64 rows
  const int n0 = nblk * 16;
  const int wv = tid >> 5;

#if __has_builtin(__builtin_amdgcn_tensor_load_to_lds)
  if (wv == 0) {
    // Tensor DMA descriptor (ISA ch.8): 2-D tile, 16 rows x 500 elems, f32.
    unsigned long ga = (unsigned long)(WhhP + (long)n0 * N_HID);
    unsigned lds_off = (unsigned)(unsigned long)(void*)ldsB;  // LDS aperture: offset in addr[31:0]
    u32x4 g0;
    g0[0] = 1u;                                   // count=1, no gather
    g0[1] = lds_off;                              // lds_addr (bytes)
    g0[2] = (unsigned)ga;                         // global_addr[95:64]
    g0[3] = ((unsigned)(ga >> 32) & 0x01FFFFFFu)  // global_addr[120:96]
            | 0x80000000u;                        // type=2 ("image") in bits 127:126
    i32x8 g1;
    g1[0] = 2 << 16;                              // workgroup_mask=0, data_size=2 (4B), no flags
    g1[1] = (N_HID & 0xFFFF) << 16;               // tensor_dim0[15:0] @ bit48
    g1[2] = 16 << 16;                             // tensor_dim0[31:16]=0 | tensor_dim1=16 @ bit80
    g1[3] = (N_HID & 0xFFFF) << 16;               // tensor_dim1[31:16]=0 | tile_dim0=500 @ bit112
    g1[4] = 16;                                   // tile_dim1=16, tile_dim2=0
    g1[5] = N_HID;                                // tensor_dim0_stride[31:0]=500
    g1[6] = 0;                                    // stride0[47:32]=0, tensor_dim1_stride lo=0
    g1[7] = 0;
    i32x4 gz4 = {0, 0, 0, 0};                     // groups 2/3 unused (2-D tensor)
    i32x8 gz8 = {0, 0, 0, 0, 0, 0, 0, 0};         // extra group (6-arg toolchain form)
    __builtin_amdgcn_tensor_load_to_lds(g0, g1, gz4, gz4, gz8, 0);
    __builtin_amdgcn_s_wait_tensorcnt(0);
  }
  __syncthreads();
#else
  for (int idx = tid; idx < 16 * (N_HID / 2); idx += 128) {
    int c5 = idx & 15, kp = idx >> 4;
    v2f w = *(const v2f*)(WhhP + (long)(n0 + c5) * N_HID + 2 * kp);
    *(v2f*)(ldsB + c5 * N_HID + 2 * kp) = w;
  }
  __syncthreads();
#endif

  const float* Hp;
  long ldh;
  if (t == 0) { Hp = h0;                             ldh = N_HID; }
  else        { Hp = hlist + (long)(t - 1) * N_HID;  ldh = (long)LENGTH * N_HID; }

  const int m0   = mblk * 64 + wv * 16;
  const int lane = tid & 31;
  const int half = lane >> 4, l16 = lane & 15;
  const float* Ap = Hp + (long)(m0 + l16) * ldh + 2 * half;
  const float* Bp = ldsB + l16 * N_HID + 2 * half;   // row-major TDM tile

  v8f c = {};
#pragma unroll 5
  for (int k = 0; k < N_HID; k += 4) {
    v2f a = *(const v2f*)(Ap + k);
    v2f b = *(const v2f*)(Bp + k);
    c = WMMA_F32(a, b, c);
  }

#pragma unroll
  for (int j = 0; j < 8; ++j) {
    int gr = m0 + j + 8 * half;           // batch index
    int gc = n0 + l16;                    // hidden index
    if (gc < N_HID) {
      float u   = U[((long)gr * LENGTH + t) * HPAD + gc];
      float hp  = Hp[(long)gr * ldh + gc];
      float al  = alpha[gc];
      float pre = c[j] + u;
      float hn  = (1.f - al) * hp + al * fmaxf(pre, 0.f);
      hlist[(long)gr * LENGTH * N_HID + (long)t * N_HID + gc] = hn;
    }
  }
}

// ---------------------------------------------------------------------------
// output_list[r, :] = hidden_list[r, :] @ W_out^T + b_out   (rows r contiguous)
// ---------------------------------------------------------------------------
__global__ void kOut(const float* __restrict__ hlist, const float* __restrict__ Wout,
                     const float* __restrict__ bout, float* __restrict__ olist) {
  const int wave  = blockIdx.x * 4 + (threadIdx.x >> 5);
  const int mtile = wave >> 2;        // 8000 mtiles
  const int ntile = wave & 3;         // 4 ntiles (N_OUT/16)
  const int m0 = mtile * 16, n0 = ntile * 16;
  __builtin_prefetch(hlist + (long)m0 * N_HID, 0, 0);
  v8f c = wmma_tile_nc(hlist, N_HID, Wout, N_HID, m0, n0, N_HID);
  const int lane = threadIdx.x & 31;
  const int half = lane >> 4, l16 = lane & 15;
#pragma unroll
  for (int j = 0; j < 8; ++j)
    olist[(long)(m0 + j + 8 * half) * N_OUT + (n0 + l16)] = c[j] + bout[n0 + l16];
}

// final_h[b, h] = hidden_list[b, LENGTH-1, h]
__global__ void kFinal(const float* __restrict__ hlist, float* __restrict__ hfin) {
  int i = blockIdx.x * blockDim.x + threadIdx.x;
  if (i < BATCH * N_HID) {
    int b = i / N_HID, h = i - b * N_HID;
    hfin[i] = hlist[(long)b * LENGTH * N_HID + (long)(LENGTH - 1) * N_HID + h];
  }
}

// ---------------------------------------------------------------------------
extern "C" void kernel_launch(void* const* d_in, const int* in_sizes, int n_in,
                              void* d_out, int out_size, void* d_ws, size_t ws_size,
                              hipStream_t stream) {
  const float* x     = (const float*)d_in[0];
  const float* h0    = (const float*)d_in[1];
  const float* Win   = (const float*)d_in[2];
  const float* bin   = (const float*)d_in[3];
  const float* Wih   = (const float*)d_in[4];
  const float* bih   = (const float*)d_in[5];
  const float* Whh   = (const float*)d_in[6];
  const float* bhh   = (const float*)d_in[7];
  const float* Wout  = (const float*)d_in[8];
  const float* bout  = (const float*)d_in[9];
  const float* alpha = (const float*)d_in[10];

  float* hlist = (float*)d_out;                              // [256, 500, 500]
  float* olist = hlist + (long)BATCH * LENGTH * N_HID;       // [256, 500, 64]
  float* hfin  = olist + (long)BATCH * LENGTH * N_OUT;       // [256, 500]

  float* U    = (float*)d_ws;                                // [128000, 512]
  float* Wc   = U + (long)MROWS * HPAD;                      // [512, 128]
  float* bc   = Wc + (long)HPAD * N_IN;                      // [512]
  float* WhhP = bc + HPAD;                                   // [512, 500]

  // Fold input path: W_comb = W_ih @ W_in, bias_comb = W_ih@b_in + b_ih + b_hh
  kCombW<<<64, 128, 0, stream>>>(Wih, Win, Wc);
  kBias<<<1, HPAD, 0, stream>>>(Wih, bin, bih, bhh, bc);
  kPadW<<<((HPAD * N_HID) + 255) / 256, 256, 0, stream>>>(Whh, WhhP);

  // Batch-parallel input projection for all (b, t)
  kU<<<(MROWS / 16) * (HPAD / 16) / 4, 128, 0, stream>>>(x, Wc, bc, U);

  // Sequential recurrence: 500 small GEMM+elementwise steps
  for (int t = 0; t < LENGTH; ++t)
    kStep<<<4 * 32, 128, 0, stream>>>(h0, hlist, U, WhhP, alpha, t);

  // Batch-parallel output projection + final hidden copy
  kOut<<<(MROWS / 16) * (N_OUT / 16) / 4, 128, 0, stream>>>(hlist, Wout, bout, olist);
  kFinal<<<(BATCH * N_HID + 255) / 256, 256, 0, stream>>>(hlist, hfin);
}